// fpsModule_16561393893902
// MI455X (gfx1250) — compile-verified
//
#include <hip/hip_runtime.h>

#define NT 1024  // 32 wave32s = one full WGP workgroup
#define NT_LOG2 10

// ---- CDNA5 async global->LDS path (guarded; falls back to plain loads) ----
#if defined(__gfx1250__) && __has_builtin(__builtin_amdgcn_global_load_async_to_lds_b32)
#define FPS_HAVE_ASYNC 1
#else
#define FPS_HAVE_ASYNC 0
#endif

__device__ __forceinline__ void fps_wait_async0() {
#if __has_builtin(__builtin_amdgcn_s_wait_asynccnt)
  __builtin_amdgcn_s_wait_asynccnt(0);
#else
  asm volatile("s_wait_asynccnt 0" ::: "memory");
#endif
}

// Copy 3 consecutive floats (one point) from global into an LDS slot.
// Executed by a single lane (lane 0 of wave 0).
__device__ __forceinline__ void fps_fetch_q(const float* src, float* qbuf) {
#if FPS_HAVE_ASYNC
  auto g = (__attribute__((address_space(1))) int*)(src);
  auto l = (__attribute__((address_space(3))) int*)(qbuf);
  // INST_OFFSET applies to both the global and LDS address (ISA 08 §4.4).
  __builtin_amdgcn_global_load_async_to_lds_b32(g, l, 0, 0);
  __builtin_amdgcn_global_load_async_to_lds_b32(g, l, 4, 0);
  __builtin_amdgcn_global_load_async_to_lds_b32(g, l, 8, 0);
  fps_wait_async0();
#else
  qbuf[0] = src[0];
  qbuf[1] = src[1];
  qbuf[2] = src[2];
#endif
}

// Persistent FPS kernel: one workgroup per batch, full sampling loop inside.
// LDS layout (dynamic): dist[n] | redV[32] | redI[32] | qbuf[4]
// Requires n % NT == 0 (n = 65536 here).
__global__ __launch_bounds__(NT, 1)
void fps_persist_kernel(const float* __restrict__ pts, int n, int npoints,
                        int* __restrict__ out) {
  extern __shared__ float smem[];
  float* dist = smem;                    // n floats (256 KB) — running min-dist
  float* redV = smem + n;                // 32 per-wave best values
  int*   redI = (int*)(redV + 32);       // 32 per-wave best indices
  float* qbuf = (float*)(redI + 32);     // broadcast slot for last point coords

  const int tid   = threadIdx.x;
  const int iters = n >> NT_LOG2;        // uniform trip count: no exec-mask churn
  const float* P = pts + (size_t)blockIdx.x * (size_t)n * 3;
  int* O = out + (size_t)blockIdx.x * (size_t)npoints;

  // Per-thread bases hoisted once: one 64-bit VGPR pair + immediate offsets
  // inside the unrolled hot loop (chunk stride 12288 B fits the 24-bit IOFFSET).
  const float* Pt = P + 3 * (size_t)tid;   // this thread's coordinate stream
  float* dt = dist + tid;                  // this thread's dist slots (stride NT)

  // Init dist buffer; prime L2/WGP$ with the coordinate stream.
  for (int k = 0; k < iters; ++k) {
    dt[(size_t)k << NT_LOG2] = 1.0e10f;
    __builtin_prefetch(Pt + (size_t)k * (3 * NT), 0, 3);  // -> global_prefetch_b8
  }
  if (tid == 0) {
    O[0] = 0;                 // first sample is index 0 (reference semantics)
    fps_fetch_q(P, qbuf);     // q = point 0
  }
  __syncthreads();

  for (int s = 1; s < npoints; ++s) {
    const float qx = qbuf[0];
    const float qy = qbuf[1];
    const float qz = qbuf[2];

    float bv = -1.0f;  // all distances are >= 0
    int   bk = 0;      // best chunk; ascending k => first-max tie-break
#pragma unroll 4
    for (int k = 0; k < iters; ++k) {
      const size_t co = (size_t)k * (3 * NT);
      const float x  = Pt[co + 0];
      const float y  = Pt[co + 1];
      const float z  = Pt[co + 2];
      const float dx = x - qx, dy = y - qy, dz = z - qz;
      const float d  = dx * dx + dy * dy + dz * dz;
      const float nd = fminf(dt[(size_t)k << NT_LOG2], d);
      dt[(size_t)k << NT_LOG2] = nd;
      if (nd > bv) { bv = nd; bk = k; }
    }
    int bi = tid + (bk << NT_LOG2);  // reconstruct global index once

    // Intra-wave argmax (wave32 butterfly); smaller index wins ties.
#pragma unroll
    for (int m = 16; m > 0; m >>= 1) {
      const float ov = __shfl_xor(bv, m, 32);
      const int   oi = __shfl_xor(bi, m, 32);
      if (ov > bv || (ov == bv && oi < bi)) { bv = ov; bi = oi; }
    }
    if ((tid & 31) == 0) { redV[tid >> 5] = bv; redI[tid >> 5] = bi; }
    __syncthreads();

    // Cross-wave argmax in wave 0 (exactly threads 0..31 on wave32).
    if (tid < 32) {
      bv = redV[tid];
      bi = redI[tid];
#pragma unroll
      for (int m = 16; m > 0; m >>= 1) {
        const float ov = __shfl_xor(bv, m, 32);
        const int   oi = __shfl_xor(bi, m, 32);
        if (ov > bv || (ov == bv && oi < bi)) { bv = ov; bi = oi; }
      }
      if (tid == 0) {
        O[s] = bi;
        // Async-DMA the newly selected point into the LDS broadcast slot;
        // visible to all waves after the barrier below.
        fps_fetch_q(P + 3 * (size_t)bi, qbuf);
      }
    }
    __syncthreads();
  }
}

extern "C" void kernel_launch(void* const* d_in, const int* in_sizes, int n_in,
                              void* d_out, int out_size, void* d_ws, size_t ws_size,
                              hipStream_t stream) {
  (void)n_in; (void)d_ws; (void)ws_size;
  // setup_inputs order: d_in[0] = npoints (scalar), d_in[1] = t_in (16,65536,3) f32
  const float* t_in = (const float*)d_in[1];
  int* out = (int*)d_out;  // output dtype is int32; write raw int32 bits

  const int B = 16;                         // batch dim per reference shape
  const int n = in_sizes[1] / (B * 3);      // 65536
  const int npoints = out_size / B;         // 2048

  const size_t shmem = (size_t)n * sizeof(float)   // dist
                     + 32 * sizeof(float)          // redV
                     + 32 * sizeof(int)            // redI
                     + 4 * sizeof(float);          // qbuf

  // Opt-in to >64KB dynamic LDS (no-op / harmless if unsupported).
  (void)hipFuncSetAttribute((const void*)fps_persist_kernel,
                            hipFuncAttributeMaxDynamicSharedMemorySize,
                            (int)shmem);

  fps_persist_kernel<<<dim3(B), dim3(NT), shmem, stream>>>(t_in, n, npoints, out);
}